// PickPAtDCA_21784074125452
// MI455X (gfx1250) — compile-verified
//
#include <hip/hip_runtime.h>
#include <hip/hip_bf16.h>

// PickPAtDCA for MI455X (gfx1250, wave32).
// Memory-bound streaming segmented reduction:
//   traffic ~ N*(64+4+4) B = 288 MB  ->  ~12.4 us floor at 23.3 TB/s.
// h rows are DMA-staged into LDS with GLOBAL_LOAD_ASYNC_TO_LDS_B128
// (double-buffered, ASYNCcnt-tracked), per-lane private slots -> no barriers.
// ws layout: [ u64 packed_min[B] ][ float sums[B][4] = {sx,sy,sz,sE} ]

#define TPB 256        // 8 wave32 per block
#define MAX_BLOCKS 2048
#define ROWF 14        // floats per LDS row: 12 staged (48B) + 2 pad -> 56B stride

__device__ __forceinline__ unsigned long long u64min(unsigned long long a,
                                                     unsigned long long b) {
  return a < b ? a : b;
}

// Stage one 48B h-row slice (floats 0..11) global->LDS via CDNA5 async DMA.
// INST_OFFSET is added to BOTH the global and LDS addresses (ISA 10.x async),
// so one LDS-base VGPR + one 64-bit address VGPR pair cover all three b128s.
__device__ __forceinline__ void stage_row_async(const float* gsrc, float* lds_row) {
  unsigned lp = (unsigned)(size_t)lds_row;           // low 32 bits = LDS offset
  unsigned long long ga = (unsigned long long)(size_t)gsrc;
  asm volatile(
      "global_load_async_to_lds_b128 %0, %1, off\n\t"
      "global_load_async_to_lds_b128 %0, %1, off offset:16\n\t"
      "global_load_async_to_lds_b128 %0, %1, off offset:32"
      :: "v"(lp), "v"(ga)
      : "memory");
}

// ---------------------------------------------------------------- init ws ---
__global__ void __launch_bounds__(256)
pp_init_ws(unsigned long long* __restrict__ packed_min,
           float4* __restrict__ sums4, int nb) {
  int b = blockIdx.x * blockDim.x + threadIdx.x;
  if (b < nb) {
    // upper 32 = bits(+inf) => "no track seen"; lower irrelevant.
    packed_min[b] = (0x7F800000ull << 32) | 0xFFFFFFFFull;
    sums4[b] = make_float4(0.f, 0.f, 0.f, 0.f);
  }
}

// ----------------------------------------------------------- main streaming --
__global__ void __launch_bounds__(TPB)
pp_scan(const float* __restrict__ h,
        const float* __restrict__ chi,
        const int* __restrict__ bidx,
        long long n,
        unsigned long long* __restrict__ packed_min,
        float* __restrict__ sums) {
  __shared__ float tbuf[2][TPB * ROWF];   // 2 x 14 KB double buffer

  const int tid  = threadIdx.x;
  const int lane = tid & 31;

  // contiguous chunk per block keeps batch ids locally clustered (sorted input)
  const long long per   = (n + (long long)gridDim.x - 1) / (long long)gridDim.x;
  const long long start = (long long)blockIdx.x * per;
  long long end = start + per;
  if (end > n) end = n;

  // wave-uniform accumulator (identical registers in all 32 lanes)
  int accB = -1;
  float aE = 0.f, ax = 0.f, ay = 0.f, az = 0.f;
  unsigned long long amin = ~0ull;

  // prologue: stage tile 0 (addresses clamped so every wave issues uniformly)
  if (start < end) {
    long long i0 = start + tid;
    if (i0 > n - 1) i0 = n - 1;
    stage_row_async(h + i0 * 16, &tbuf[0][tid * ROWF]);
  }

  int cur = 0;
  for (long long it = start; it < end; it += TPB, cur ^= 1) {
    const long long i = it + tid;
    const bool valid = (i < end);
    const bool has_next = (it + TPB) < end;   // uniform across block

    if (has_next) {
      long long i2 = it + TPB + tid;
      if (i2 > n - 1) i2 = n - 1;             // clamp: uniform issue, safe addr
      stage_row_async(h + i2 * 16, &tbuf[cur ^ 1][tid * ROWF]);
      // async loads complete in order: <=3 outstanding => tile t's 3 are done
      asm volatile("s_wait_asynccnt 0x3" ::: "memory");
    } else {
      asm volatile("s_wait_asynccnt 0x0" ::: "memory");
    }

    int b = -1;
    float wE = 0.f, wx = 0.f, wy = 0.f, wz = 0.f;
    unsigned long long pk = ~0ull;

    if (valid) {
      // prefetch the small side-streams ahead (global_prefetch_b8)
      if (it + (long long)TPB * 4 < end) {
        __builtin_prefetch(chi + i + (long long)TPB * 4, 0, 0);
        __builtin_prefetch(bidx + i + (long long)TPB * 4, 0, 0);
      }

      // read this lane's staged row: 56B stride = 14 dwords -> 32 distinct
      // LDS banks across the wave; float2 keeps 8B alignment.
      const float2* r = (const float2*)&tbuf[cur][tid * ROWF];
      const float2 xy  = r[0];                // x, y
      const float2 zh3 = r[1];                // z, h3
      const float2 h45 = r[2];                // h4, h5
      const float2 h67 = r[3];                // h6, h7
      const float  E   = tbuf[cur][tid * ROWF + 8];

      const float c = chi[i];
      b = bidx[i];

      // argmax(h[:,3:7]) == 1 with first-occurrence tie-break:
      // h4 strictly beats h3, and is >= h5, h6.
      const bool is_track = (h45.x > zh3.y) & (h45.x >= h45.y) & (h45.x >= h67.x);
      const unsigned cb = is_track ? __float_as_uint(c) : 0x7F800000u;
      pk = ((unsigned long long)cb << 32) | (unsigned)(unsigned int)i;

      wE = E;
      wx = xy.x * E;
      wy = xy.y * E;
      wz = zh3.x * E;
    }

    const int b0 = __shfl(b, 0);
    const bool uniform = __all(valid && (b == b0));

    if (uniform) {
      // wave32 butterfly: sums + u64 min; all lanes end with the wave total
#pragma unroll
      for (int off = 16; off >= 1; off >>= 1) {
        wE += __shfl_xor(wE, off);
        wx += __shfl_xor(wx, off);
        wy += __shfl_xor(wy, off);
        wz += __shfl_xor(wz, off);
        pk = u64min(pk, __shfl_xor(pk, off));
      }
      if (b0 != accB) {
        if (accB >= 0 && lane == 0) {  // flush previous graph (rare)
          atomicAdd(&sums[accB * 4 + 0], ax);
          atomicAdd(&sums[accB * 4 + 1], ay);
          atomicAdd(&sums[accB * 4 + 2], az);
          atomicAdd(&sums[accB * 4 + 3], aE);
          atomicMin(&packed_min[accB], amin);
        }
        accB = b0; ax = wx; ay = wy; az = wz; aE = wE; amin = pk;
      } else {
        ax += wx; ay += wy; az += wz; aE += wE;
        amin = u64min(amin, pk);
      }
    } else if (valid) {
      // mixed wave (~1.6% of iterations): direct commutative atomics
      atomicAdd(&sums[b * 4 + 0], wx);
      atomicAdd(&sums[b * 4 + 1], wy);
      atomicAdd(&sums[b * 4 + 2], wz);
      atomicAdd(&sums[b * 4 + 3], wE);
      atomicMin(&packed_min[b], pk);
    }
  }

  if (accB >= 0 && lane == 0) {
    atomicAdd(&sums[accB * 4 + 0], ax);
    atomicAdd(&sums[accB * 4 + 1], ay);
    atomicAdd(&sums[accB * 4 + 2], az);
    atomicAdd(&sums[accB * 4 + 3], aE);
    atomicMin(&packed_min[accB], amin);
  }
}

// ---------------------------------------------------------------- finalize ---
__global__ void __launch_bounds__(256)
pp_finalize(const float* __restrict__ h,
            const float* __restrict__ pxpypz,
            const unsigned long long* __restrict__ packed_min,
            const float* __restrict__ sums,
            float* __restrict__ out,
            long long n, int nb) {
  int b = blockIdx.x * blockDim.x + threadIdx.x;
  if (b >= nb) return;

  const unsigned long long pk = packed_min[b];
  const unsigned hi = (unsigned)(pk >> 32);
  long long idx = (hi == 0x7F800000u) ? (n - 1)           // no track / empty
                                      : (long long)(unsigned)pk;
  if (idx > n - 1) idx = n - 1;

  const float px = pxpypz[idx * 3 + 0];
  const float py = pxpypz[idx * 3 + 1];
  const float pz = pxpypz[idx * 3 + 2];
  const float hx = h[idx * 16 + 0];
  const float hy = h[idx * 16 + 1];
  const float hz = h[idx * 16 + 2];

  const float sx = sums[b * 4 + 0];
  const float sy = sums[b * 4 + 1];
  const float sz = sums[b * 4 + 2];
  const float sE = sums[b * 4 + 3];

  out[b] = sqrtf(px * px + py * py + pz * pz);     // p_tracks      [B]
  float* pdir = out + nb;                          // p_direction   [B,3]
  pdir[b * 3 + 0] = px;
  pdir[b * 3 + 1] = py;
  pdir[b * 3 + 2] = pz;
  float* dbar = out + (long long)nb * 4;           // bary - p_xyz  [B,3]
  dbar[b * 3 + 0] = sx / sE - hx;
  dbar[b * 3 + 1] = sy / sE - hy;
  dbar[b * 3 + 2] = sz / sE - hz;
}

// ------------------------------------------------------------------ launch ---
extern "C" void kernel_launch(void* const* d_in, const int* in_sizes, int n_in,
                              void* d_out, int out_size, void* d_ws, size_t ws_size,
                              hipStream_t stream) {
  const float* h      = (const float*)d_in[0];   // N x 16
  const float* pxpypz = (const float*)d_in[1];   // N x 3
  const float* chi    = (const float*)d_in[2];   // N
  const int*   bidx   = (const int*)d_in[3];     // N (sorted)
  // d_in[4] = num_graphs (device scalar) — derive on host instead:
  const long long n  = (long long)in_sizes[0] / 16;
  const int       nb = out_size / 7;             // out = B + 3B + 3B

  unsigned long long* packed_min = (unsigned long long*)d_ws;
  float*              sums       = (float*)(packed_min + nb);

  pp_init_ws<<<(nb + 255) / 256, 256, 0, stream>>>(packed_min, (float4*)sums, nb);

  long long want = (n + TPB - 1) / TPB;
  int nblocks = (int)(want < (long long)MAX_BLOCKS ? want : (long long)MAX_BLOCKS);
  if (nblocks < 1) nblocks = 1;
  pp_scan<<<nblocks, TPB, 0, stream>>>(h, chi, bidx, n, packed_min, sums);

  pp_finalize<<<(nb + 255) / 256, 256, 0, stream>>>(h, pxpypz, packed_min, sums,
                                                    (float*)d_out, n, nb);
}